// PointSegAux_88819923681863
// MI455X (gfx1250) — compile-verified
//
#include <hip/hip_runtime.h>
#include <hip/hip_bf16.h>

typedef __attribute__((ext_vector_type(2))) float v2f;
typedef __attribute__((ext_vector_type(8))) float v8f;

#define BN       16384   // B*N
#define NPTS     8192    // N per batch
#define FEAT_DIM 320
#define HID      64
#define KCHUNK   80      // K-rows of w_fc staged in LDS per pass (320/80 = 4 passes)
#define LWS      (KCHUNK + 2)  // padded LDS stride (even -> 8B-aligned b64 loads)
#define NN_EPS   1e-8f

// ---------------------------------------------------------------------------
// Stage A: fused three_nn + three_interpolate for one scale.
// One thread per query point; candidate xyz tiled through LDS (3 KB/block).
// Writes its C columns into the concatenated feature matrix X[BN,320].
// ---------------------------------------------------------------------------
template <int C>
__global__ __launch_bounds__(256) void interp_kernel(
    const float* __restrict__ points,  // [B,N,3]
    const float* __restrict__ xyz,     // [B,M,3]
    const float* __restrict__ feat,    // [B,M,C]
    float* __restrict__ X,             // [BN,320]
    int M, int colOff)
{
    __shared__ float sx[256], sy[256], sz[256];

    const int t = blockIdx.x * 256 + threadIdx.x;  // global query id
    const int b = t / NPTS;                        // whole block shares b (N%256==0)
    const float qx = points[t * 3 + 0];
    const float qy = points[t * 3 + 1];
    const float qz = points[t * 3 + 2];
    const float* kx = xyz + (size_t)b * M * 3;

    float d0 = 1e30f, d1 = 1e30f, d2 = 1e30f;
    int   i0 = 0, i1 = 0, i2 = 0;

    for (int base = 0; base < M; base += 256) {    // M is a multiple of 256
        const int j = base + threadIdx.x;
        sx[threadIdx.x] = kx[j * 3 + 0];
        sy[threadIdx.x] = kx[j * 3 + 1];
        sz[threadIdx.x] = kx[j * 3 + 2];
        __syncthreads();
#pragma unroll 4
        for (int s = 0; s < 256; ++s) {
            const float dx = qx - sx[s], dy = qy - sy[s], dz = qz - sz[s];
            const float d  = dx * dx + dy * dy + dz * dz;
            const int   gi = base + s;
            if (d < d2) {
                if (d < d1) {
                    if (d < d0) { d2 = d1; i2 = i1; d1 = d0; i1 = i0; d0 = d; i0 = gi; }
                    else        { d2 = d1; i2 = i1; d1 = d;  i1 = gi; }
                } else          { d2 = d;  i2 = gi; }
            }
        }
        __syncthreads();
    }

    const float r0 = 1.0f / (d0 + NN_EPS);
    const float r1 = 1.0f / (d1 + NN_EPS);
    const float r2 = 1.0f / (d2 + NN_EPS);
    const float rn = 1.0f / (r0 + r1 + r2);
    const float w0 = r0 * rn, w1 = r1 * rn, w2 = r2 * rn;

    const float4* f0 = (const float4*)(feat + ((size_t)b * M + i0) * C);
    const float4* f1 = (const float4*)(feat + ((size_t)b * M + i1) * C);
    const float4* f2 = (const float4*)(feat + ((size_t)b * M + i2) * C);
    float4* xo = (float4*)(X + (size_t)t * FEAT_DIM + colOff);
#pragma unroll
    for (int c = 0; c < C / 4; ++c) {
        const float4 a = f0[c], u = f1[c], v = f2[c];
        float4 o;
        o.x = w0 * a.x + w1 * u.x + w2 * v.x;
        o.y = w0 * a.y + w1 * u.y + w2 * v.y;
        o.z = w0 * a.z + w1 * u.z + w2 * v.z;
        o.w = w0 * a.w + w1 * u.w + w2 * v.w;
        xo[c] = o;
    }
}

// ---------------------------------------------------------------------------
// Stage B: out = (X @ w_fc) @ w_cls, fused.
// Per wave: 16 output rows, full 64-wide hidden dim via 4 fp32 WMMA
// accumulators, K loop in steps of 4 (V_WMMA_F32_16X16X4_F32).
// w_fc staged transposed in LDS; w_cls contraction via width-16 shfl_xor.
// ---------------------------------------------------------------------------
__global__ __launch_bounds__(256) void fc_wmma_kernel(
    const float* __restrict__ X,     // [BN,320]
    const float* __restrict__ wfc,   // [320,64]
    const float* __restrict__ wcls,  // [64]
    float* __restrict__ out)         // [BN]
{
    __shared__ float lw[HID * LWS];  // w_fc chunk, transposed [col][k]

    const int tid  = threadIdx.x;
    const int lane = tid & 31;
    const int wv   = tid >> 5;                       // 8 waves / block
    const int row0 = (blockIdx.x * 8 + wv) * 16;
    const int mrow = row0 + (lane & 15);             // A row owned by this lane
    const int kb   = (lane >> 4) * 2;                // K-pair: 0,1 or 2,3
    const int col  = lane & 15;                      // B/D column (per tile)

    v8f acc0 = {0, 0, 0, 0, 0, 0, 0, 0};
    v8f acc1 = acc0, acc2 = acc0, acc3 = acc0;

    for (int kc = 0; kc < FEAT_DIM; kc += KCHUNK) {
        __syncthreads();
        for (int i = tid; i < KCHUNK * HID; i += 256) {
            const int k = i >> 6, c = i & 63;
            lw[c * LWS + k] = wfc[(kc + k) * HID + c];
        }
        __syncthreads();

        const float* xr = X + (size_t)mrow * FEAT_DIM + kc + kb;
        for (int k0 = 0; k0 < KCHUNK; k0 += 4) {
            const v2f a  = *(const v2f*)(xr + k0);
            const v2f b0 = *(const v2f*)(lw + (col     ) * LWS + k0 + kb);
            const v2f b1 = *(const v2f*)(lw + (col + 16) * LWS + k0 + kb);
            const v2f b2 = *(const v2f*)(lw + (col + 32) * LWS + k0 + kb);
            const v2f b3 = *(const v2f*)(lw + (col + 48) * LWS + k0 + kb);
            acc0 = __builtin_amdgcn_wmma_f32_16x16x4_f32(false, a, false, b0, (short)0, acc0, false, false);
            acc1 = __builtin_amdgcn_wmma_f32_16x16x4_f32(false, a, false, b1, (short)0, acc1, false, false);
            acc2 = __builtin_amdgcn_wmma_f32_16x16x4_f32(false, a, false, b2, (short)0, acc2, false, false);
            acc3 = __builtin_amdgcn_wmma_f32_16x16x4_f32(false, a, false, b3, (short)0, acc3, false, false);
        }
    }

    // Fused w_cls contraction: p[r] = sum_col D[r][col] * wcls[col]
    const float wc0 = wcls[col], wc1 = wcls[col + 16];
    const float wc2 = wcls[col + 32], wc3 = wcls[col + 48];
    float p[8];
#pragma unroll
    for (int r = 0; r < 8; ++r)
        p[r] = acc0[r] * wc0 + acc1[r] * wc1 + acc2[r] * wc2 + acc3[r] * wc3;
#pragma unroll
    for (int r = 0; r < 8; ++r)
        for (int off = 8; off >= 1; off >>= 1)
            p[r] += __shfl_xor(p[r], off, 16);       // reduce over 16 columns

    if ((lane & 15) == 0) {
        const int rbase = row0 + (lane >> 4) * 8;    // rows 0-7 / 8-15 halves
#pragma unroll
        for (int r = 0; r < 8; ++r) out[rbase + r] = p[r];
    }
}

// ---------------------------------------------------------------------------
extern "C" void kernel_launch(void* const* d_in, const int* in_sizes, int n_in,
                              void* d_out, int out_size, void* d_ws, size_t ws_size,
                              hipStream_t stream) {
    const float* points = (const float*)d_in[0];
    const float* xyz2   = (const float*)d_in[1];
    const float* feat2  = (const float*)d_in[2];
    const float* xyz3   = (const float*)d_in[3];
    const float* feat3  = (const float*)d_in[4];
    const float* xyz4   = (const float*)d_in[5];
    const float* feat4  = (const float*)d_in[6];
    const float* wfc    = (const float*)d_in[7];
    const float* wcls   = (const float*)d_in[8];
    float* out = (float*)d_out;
    float* X   = (float*)d_ws;  // [16384,320] fp32 = 20.97 MB scratch

    dim3 blk(256), grd(BN / 256);
    interp_kernel<64>  <<<grd, blk, 0, stream>>>(points, xyz2, feat2, X, 8192,   0);
    interp_kernel<128> <<<grd, blk, 0, stream>>>(points, xyz3, feat3, X, 4096,  64);
    interp_kernel<128> <<<grd, blk, 0, stream>>>(points, xyz4, feat4, X, 2048, 192);

    fc_wmma_kernel<<<dim3(BN / (8 * 16)), blk, 0, stream>>>(X, wfc, wcls, out);
}